// GNNConv_18399639896341
// MI455X (gfx1250) — compile-verified
//
#include <hip/hip_runtime.h>

typedef __attribute__((ext_vector_type(2))) float v2f;
typedef __attribute__((ext_vector_type(8))) float v8f;

#define D 128
#define LDS_STRIDE 132   // 16-row fp32 tile padded: bank = (4*row + k) % 64 -> conflict-free
#define LN_EPS 1e-5f

// ---------------------------------------------------------------- init
__global__ void gnn_init_kernel(float* __restrict__ deg, float* __restrict__ agg,
                                long long total, int N) {
  long long i = (long long)blockIdx.x * blockDim.x + threadIdx.x;
  if (i < total) agg[i] = 0.0f;
  if (i < N) deg[i] = 1.0f;   // self-loop contribution to degree
}

// ---------------------------------------------------------------- degree
__global__ void gnn_degree_kernel(const long long* __restrict__ ei,
                                  float* __restrict__ deg, long long E) {
  long long e = (long long)blockIdx.x * blockDim.x + threadIdx.x;
  if (e < E) atomicAdd(&deg[ei[E + e]], 1.0f);   // dst row of edge_index
}

// ---------------------------------------------------------------- rsqrt (in place: deg -> dinv)
__global__ void gnn_rsqrt_kernel(float* __restrict__ deg, int N) {
  int i = blockIdx.x * blockDim.x + threadIdx.x;
  if (i < N) deg[i] = rsqrtf(deg[i]);            // deg >= 1 always (self loops)
}

// ---------------------------------------------------------------- dual GEMM: xw = x@W, xl = x@Wl
// One block = 16 rows of x. 8 waves, wave w computes the 16x16 tile at cols [16w, 16w+16)
// for BOTH weight matrices, accumulating K=128 in 32 steps of V_WMMA_F32_16X16X4_F32.
__global__ void __launch_bounds__(256)
gnn_gemm_dual_kernel(const float* __restrict__ x,
                     const float* __restrict__ W,
                     const float* __restrict__ Wl,
                     float* __restrict__ xw,
                     float* __restrict__ xl,
                     int N) {
  __shared__ float tileA[16 * LDS_STRIDE];

  const int block_row = blockIdx.x * 16;
  const int tid  = threadIdx.x;
  const int wave = tid >> 5;        // 0..7  -> output col tile
  const int lane = tid & 31;

  // Stage 16x128 x-tile into LDS: 256 threads x 8 floats (two float4s).
  {
    const int r  = tid >> 4;        // 0..15
    const int c0 = (tid & 15) * 8;  // 0..120
    float4 a = make_float4(0.f, 0.f, 0.f, 0.f);
    float4 bq = make_float4(0.f, 0.f, 0.f, 0.f);
    if (block_row + r < N) {
      const float* src = x + (size_t)(block_row + r) * D + c0;
      a  = *(const float4*)(src);
      bq = *(const float4*)(src + 4);
    }
    float* dst = tileA + r * LDS_STRIDE + c0;   // 16B aligned (528B row stride)
    *(float4*)(dst)     = a;
    *(float4*)(dst + 4) = bq;
  }
  __syncthreads();

  const int col0    = wave * 16;
  const int halfsel = lane >> 4;    // 0: lanes 0-15 (K+0,K+1), 1: lanes 16-31 (K+2,K+3)
  const int lrow    = lane & 15;    // M index for A, N index for B/D

  v8f accW = {};
  v8f accL = {};
  const float* aBase = tileA + lrow * LDS_STRIDE + 2 * halfsel;

#pragma unroll 8
  for (int k0 = 0; k0 < D; k0 += 4) {
    // A 16x4: VGPR0 = K(k0+2h), VGPR1 = K(k0+2h+1) for row M=lrow
    v2f a = *(const v2f*)(aBase + k0);          // 8B-aligned LDS read

    // B 4x16: VGPR j holds row K = k0 + j (lanes 0-15) / k0 + j + 2 (lanes 16-31)
    const int kb = k0 + 2 * halfsel;
    v2f bW, bL;
    bW.x = W [(size_t)(kb + 0) * D + col0 + lrow];
    bW.y = W [(size_t)(kb + 1) * D + col0 + lrow];
    bL.x = Wl[(size_t)(kb + 0) * D + col0 + lrow];
    bL.y = Wl[(size_t)(kb + 1) * D + col0 + lrow];

    // 8 args: (neg_a, A, neg_b, B, c_mod, C, reuse_a, reuse_b)
    accW = __builtin_amdgcn_wmma_f32_16x16x4_f32(false, a, false, bW, (short)0, accW,
                                                 false, false);
    accL = __builtin_amdgcn_wmma_f32_16x16x4_f32(false, a, false, bL, (short)0, accL,
                                                 false, false);
  }

  // D 16x16 f32: VGPR i -> row M = i + 8*halfsel, col N = lrow
  const size_t base0 = (size_t)(block_row + 8 * halfsel) * D + col0 + lrow;
  if (block_row + 16 <= N) {
    // Fast path (all tiles when N % 16 == 0): branch-free, stores can clause.
#pragma unroll
    for (int i = 0; i < 8; ++i) {
      xw[base0 + (size_t)i * D] = accW[i];
    }
#pragma unroll
    for (int i = 0; i < 8; ++i) {
      xl[base0 + (size_t)i * D] = accL[i];
    }
  } else {
    // Ragged last tile.
#pragma unroll
    for (int i = 0; i < 8; ++i) {
      if (block_row + i + 8 * halfsel < N) {
        xw[base0 + (size_t)i * D] = accW[i];
        xl[base0 + (size_t)i * D] = accL[i];
      }
    }
  }
}

// ---------------------------------------------------------------- edge scatter
// 128 threads per edge (one per feature): agg[dst] += coef * xw[src]
__global__ void gnn_scatter_kernel(const long long* __restrict__ ei,
                                   const float* __restrict__ dinv,
                                   const float* __restrict__ xw,
                                   float* __restrict__ agg, long long E) {
  const long long t = (long long)blockIdx.x * blockDim.x + threadIdx.x;
  const long long e = t >> 7;
  const int f = (int)(t & (D - 1));
  if (e >= E) return;
  const long long s = ei[e];       // src
  const long long d = ei[E + e];   // dst
  const float c = dinv[s] * dinv[d];
  atomicAdd(&agg[d * D + f], xw[s * D + f] * c);
}

// ---------------------------------------------------------------- fuse + LayerNorm
// One wave per node; lane owns 4 consecutive features (coalesced float4).
__global__ void gnn_finalize_ln_kernel(const float* __restrict__ agg,
                                       const float* __restrict__ xw,
                                       const float* __restrict__ xl,   // aliases out
                                       const float* __restrict__ dinv,
                                       const float* __restrict__ b,
                                       const float* __restrict__ bl,
                                       const float* __restrict__ gamma,
                                       const float* __restrict__ beta,
                                       float* __restrict__ out, int N) {
  const int node = (int)(((long long)blockIdx.x * blockDim.x + threadIdx.x) >> 5);
  const int lane = threadIdx.x & 31;
  if (node >= N) return;

  const float di = dinv[node];
  const float cself = di * di;          // self-loop coefficient
  const size_t base = (size_t)node * D;
  const int f0 = lane * 4;

  const float4 a4  = *(const float4*)(agg + base + f0);
  const float4 w4  = *(const float4*)(xw  + base + f0);
  const float4 l4  = *(const float4*)(xl  + base + f0);
  const float4 b4  = *(const float4*)(b   + f0);
  const float4 bl4 = *(const float4*)(bl  + f0);

  float h0 = a4.x + cself * w4.x + l4.x + b4.x + bl4.x;
  float h1 = a4.y + cself * w4.y + l4.y + b4.y + bl4.y;
  float h2 = a4.z + cself * w4.z + l4.z + b4.z + bl4.z;
  float h3 = a4.w + cself * w4.w + l4.w + b4.w + bl4.w;

  float sum = h0 + h1 + h2 + h3;
#pragma unroll
  for (int m = 16; m >= 1; m >>= 1) sum += __shfl_xor(sum, m, 32);
  const float mean = sum * (1.0f / (float)D);

  const float d0 = h0 - mean, d1 = h1 - mean, d2 = h2 - mean, d3 = h3 - mean;
  float vs = d0 * d0 + d1 * d1 + d2 * d2 + d3 * d3;
#pragma unroll
  for (int m = 16; m >= 1; m >>= 1) vs += __shfl_xor(vs, m, 32);
  const float inv = rsqrtf(vs * (1.0f / (float)D) + LN_EPS);

  const float4 g4  = *(const float4*)(gamma + f0);
  const float4 be4 = *(const float4*)(beta  + f0);

  float4 o;
  o.x = d0 * inv * g4.x + be4.x;
  o.y = d1 * inv * g4.y + be4.y;
  o.z = d2 * inv * g4.z + be4.z;
  o.w = d3 * inv * g4.w + be4.w;
  *(float4*)(out + base + f0) = o;
}

// ---------------------------------------------------------------- launch
extern "C" void kernel_launch(void* const* d_in, const int* in_sizes, int n_in,
                              void* d_out, int out_size, void* d_ws, size_t ws_size,
                              hipStream_t stream) {
  const float*     x     = (const float*)d_in[0];
  const long long* ei    = (const long long*)d_in[1];   // int64 [2, E]
  const float*     W     = (const float*)d_in[2];
  const float*     b     = (const float*)d_in[3];
  const float*     Wl    = (const float*)d_in[4];
  const float*     bl    = (const float*)d_in[5];
  const float*     gamma = (const float*)d_in[6];
  const float*     beta  = (const float*)d_in[7];
  float*           out   = (float*)d_out;

  const int       N = in_sizes[0] / D;
  const long long E = (long long)in_sizes[1] / 2;

  // Workspace: deg/dinv (N, padded) | xw (N*D) | agg (N*D).  xl lives in d_out.
  float* deg = (float*)d_ws;
  float* xw  = deg + (((size_t)N + 255) & ~(size_t)255);
  float* agg = xw + (size_t)N * D;
  float* xl  = out;

  const long long total = (long long)N * D;

  gnn_init_kernel<<<(int)((total + 255) / 256), 256, 0, stream>>>(deg, agg, total, N);
  gnn_degree_kernel<<<(int)((E + 255) / 256), 256, 0, stream>>>(ei, deg, E);
  gnn_rsqrt_kernel<<<(N + 255) / 256, 256, 0, stream>>>(deg, N);
  gnn_gemm_dual_kernel<<<(N + 15) / 16, 256, 0, stream>>>(x, W, Wl, xw, xl, N);

  const long long sthreads = E * D;
  gnn_scatter_kernel<<<(int)((sthreads + 255) / 256), 256, 0, stream>>>(ei, deg, xw, agg, E);

  gnn_finalize_ln_kernel<<<(int)(((long long)N * 32 + 255) / 256), 256, 0, stream>>>(
      agg, xw, xl, deg, b, bl, gamma, beta, out, N);
}